// SeqModelFused_24618752541423
// MI455X (gfx1250) — compile-verified
//
#include <hip/hip_runtime.h>
#include <hip/hip_bf16.h>
#include <stdint.h>

// Problem constants (match reference setup_inputs)
#define T_DIM 128
#define B_DIM 64
#define N_DIM 1024               // hidden size (K and N of both GEMMs)
#define M_ROWS (T_DIM * B_DIM)   // 8192 rows

typedef __attribute__((ext_vector_type(2))) float v2f;
typedef __attribute__((ext_vector_type(4))) float v4f;
typedef __attribute__((ext_vector_type(8))) float v8f;
typedef __attribute__((ext_vector_type(4))) unsigned int u32x4;
typedef __attribute__((ext_vector_type(8))) int i32x8;
typedef __attribute__((ext_vector_type(4))) int i32x4;

// Tiling
#define BM 128
#define BN 128
#define BK 32
#define LDS_STRIDE 34            // floats per row: 32 data + 2 pad (TDM pad_interval=32dw, pad_amount=2dw)
#define TILE_FLOATS (128 * LDS_STRIDE)   // 4352 floats per staged tile

#define HAVE_TDM (__has_builtin(__builtin_amdgcn_tensor_load_to_lds) && \
                  __has_builtin(__builtin_amdgcn_s_wait_tensorcnt))

#if HAVE_TDM
// LDS byte offset of a generic pointer to a __shared__ object (addrspacecast -> ptrtoint).
__device__ __forceinline__ uint32_t lds_byte_offset(const void* p) {
    return (uint32_t)(uintptr_t)(__attribute__((address_space(3))) char*)(void*)(const_cast<void*>(p));
}

// Issue one TDM load of a 128x32 f32 tile (row stride N_DIM elements) into LDS with
// 2-DWORD padding after every 32 DWORDs (=> 34-float LDS row stride).
__device__ __forceinline__ void tdm_load_tile_128x32(const float* gsrc, uint32_t lds_addr) {
    const uint64_t ga = (uint64_t)(uintptr_t)gsrc;
    u32x4 g0;
    g0.x = 1u;                                       // count=1 valid descriptor, user mode
    g0.y = lds_addr;                                 // lds_addr [63:32]
    g0.z = (uint32_t)ga;                             // global_addr [95:64]
    g0.w = (uint32_t)((ga >> 32) & 0x01FFFFFFu)      // global_addr [120:96]
         | (2u << 30);                               // type=2 ("image") [127:126]

    i32x8 g1;
    g1[0] = (int)((2u << 16)                         // data_size = 4 bytes
                | (1u << 20)                         // pad_enable
                | (4u << 22)                         // pad_interval: 32 DWORDs
                | (1u << 25));                       // pad_amount: 2 DWORDs
    g1[1] = (int)((uint32_t)(N_DIM & 0xFFFF) << 16); // tensor_dim0[15:0] @ bits[63:48]
    g1[2] = (int)((uint32_t)(M_ROWS & 0xFFFF) << 16);// tensor_dim0[31:16]=0 | tensor_dim1[15:0]
    g1[3] = (int)((uint32_t)BK << 16);               // tensor_dim1[31:16]=0 | tile_dim0=32
    g1[4] = (int)(uint32_t)BM;                       // tile_dim1=128 | tile_dim2=0
    g1[5] = (int)(uint32_t)N_DIM;                    // tensor_dim0_stride[31:0] = 1024 elems
    g1[6] = 0;                                       // stride0 hi | tensor_dim1_stride lo
    g1[7] = 0;                                       // tensor_dim1_stride hi

    i32x4 z4 = (i32x4)0;                             // groups 2/3 unused for 2D tile
    i32x8 z8 = (i32x8)0;                             // extra group (clang-23 6-arg form), unused
    __builtin_amdgcn_tensor_load_to_lds(g0, g1, z4, z4, z8, /*cpol=*/0);
}
#endif

// C[r, m] = sum_k A[r, k] * W[m, k]   (A: Mrows x K row-major, W: N x K row-major == B^T)
// V_WMMA_F32_16X16X4_F32: A frag 16x4 (lane=M, vgpr+lanehalf=K),
// B frag 4x16 (lane=N, vgpr+lanehalf=K), C/D 16x16 f32 (8 VGPRs).
__global__ __launch_bounds__(256) void gemm_f32_wmma(const float* __restrict__ A,
                                                     const float* __restrict__ W,
                                                     float* __restrict__ C) {
    const int tid  = threadIdx.x;
    const int lane = tid & 31;
    const int wave = tid >> 5;        // 0..7
    const int wm   = wave & 1;        // 2 waves along M  -> 64 rows each
    const int wn   = wave >> 1;       // 4 waves along N  -> 32 cols each
    const int half = lane >> 4;       // lane half selects K pair / M+8 rows
    const int ln   = lane & 15;       // M (A frag), N (B frag, C cols)

    const int mBlk = blockIdx.y * BM;
    const int nBlk = blockIdx.x * BN;

    v8f acc[4][2];
#pragma unroll
    for (int tm = 0; tm < 4; ++tm)
#pragma unroll
        for (int tn = 0; tn < 2; ++tn)
            acc[tm][tn] = (v8f)0.0f;

#if HAVE_TDM
    // Double-buffered LDS: [A0 | B0 | A1 | B1], each 128x34 floats, filled by TDM DMA.
    __shared__ float lds[4 * TILE_FLOATS];
    const uint32_t ldsBase = lds_byte_offset(lds);
    const float* aBase = A + (size_t)mBlk * N_DIM;
    const float* wBase = W + (size_t)nBlk * N_DIM;

    if (wave == 0) {   // scalar branch: only wave 0 issues DMA (EXEC ignored by TDM)
        tdm_load_tile_128x32(aBase, ldsBase);
        tdm_load_tile_128x32(wBase, ldsBase + TILE_FLOATS * 4u);
    }

    const int nTiles = N_DIM / BK;    // 32
    for (int kt = 0; kt < nTiles; ++kt) {
        const int cur = kt & 1;
        const float* As = lds + cur * 2 * TILE_FLOATS;
        const float* Bs = As + TILE_FLOATS;

        if (wave == 0) {
            if (kt + 1 < nTiles) {    // prefetch next tile pair into the other buffer
                const int nxt = cur ^ 1;
                const uint32_t aOff = ldsBase + (uint32_t)(nxt * 2 * TILE_FLOATS) * 4u;
                tdm_load_tile_128x32(aBase + (kt + 1) * BK, aOff);
                tdm_load_tile_128x32(wBase + (kt + 1) * BK, aOff + TILE_FLOATS * 4u);
                __builtin_amdgcn_s_wait_tensorcnt(2);   // current pair complete
            } else {
                __builtin_amdgcn_s_wait_tensorcnt(0);
            }
        }
        __syncthreads();              // current buffer ready for all waves
#else
    __shared__ float AsBuf[BM * LDS_STRIDE];
    __shared__ float BsBuf[BN * LDS_STRIDE];
    for (int k0 = 0; k0 < N_DIM; k0 += BK) {
        const float* As = AsBuf;
        const float* Bs = BsBuf;
        // Cooperative load: 1024 float4 slots per tile, 256 threads -> 4 iterations.
#pragma unroll
        for (int i = 0; i < 4; ++i) {
            const int lin = tid + i * 256;
            const int row = lin >> 3;
            const int seg = lin & 7;
            const float* ga = A + (size_t)(mBlk + row) * N_DIM + k0 + seg * 4;
            const float* gw = W + (size_t)(nBlk + row) * N_DIM + k0 + seg * 4;
            v4f va = *(const v4f*)ga;
            v4f vw = *(const v4f*)gw;
            if (k0 + BK < N_DIM) {
                __builtin_prefetch(ga + BK, 0, 0);
                __builtin_prefetch(gw + BK, 0, 0);
            }
            float* pa = &AsBuf[row * LDS_STRIDE + seg * 4];
            float* pw = &BsBuf[row * LDS_STRIDE + seg * 4];
            *(v2f*)(pa + 0) = __builtin_shufflevector(va, va, 0, 1);
            *(v2f*)(pa + 2) = __builtin_shufflevector(va, va, 2, 3);
            *(v2f*)(pw + 0) = __builtin_shufflevector(vw, vw, 0, 1);
            *(v2f*)(pw + 2) = __builtin_shufflevector(vw, vw, 2, 3);
        }
        __syncthreads();
#endif

#pragma unroll
        for (int kk = 0; kk < BK / 4; ++kk) {
            v2f afrag[4];
#pragma unroll
            for (int tm = 0; tm < 4; ++tm) {
                const int m = wm * 64 + tm * 16 + ln;
                afrag[tm] = *(const v2f*)&As[m * LDS_STRIDE + kk * 4 + half * 2];
            }
            v2f bfrag[2];
#pragma unroll
            for (int tn = 0; tn < 2; ++tn) {
                const int n = wn * 32 + tn * 16 + ln;
                bfrag[tn] = *(const v2f*)&Bs[n * LDS_STRIDE + kk * 4 + half * 2];
            }
#pragma unroll
            for (int tm = 0; tm < 4; ++tm)
#pragma unroll
                for (int tn = 0; tn < 2; ++tn)
                    acc[tm][tn] = __builtin_amdgcn_wmma_f32_16x16x4_f32(
                        false, afrag[tm], false, bfrag[tn],
                        (short)0, acc[tm][tn], false, false);
        }
        __syncthreads();              // buffer free for reuse / next cooperative load
    }

    // Epilogue: C/D layout — VGPR v holds row M = v + half*8, col N = ln
#pragma unroll
    for (int tm = 0; tm < 4; ++tm)
#pragma unroll
        for (int tn = 0; tn < 2; ++tn) {
            const int colBase = nBlk + wn * 32 + tn * 16 + ln;
            const int rowBase = mBlk + wm * 64 + tm * 16 + half * 8;
#pragma unroll
            for (int v = 0; v < 8; ++v)
                C[(size_t)(rowBase + v) * N_DIM + colBase] = acc[tm][tn][v];
        }
}

// LIF scan over T, in place: reads h[t], writes spike[t] into the same buffer.
__global__ __launch_bounds__(256) void lif_scan(float* __restrict__ H) {
    const int idx = blockIdx.x * blockDim.x + threadIdx.x;   // over B*N = 65536
    float v = 0.0f;
#pragma unroll 4
    for (int t = 0; t < T_DIM; ++t) {
        const size_t off = (size_t)t * (B_DIM * N_DIM) + idx;
        const float h = 0.5f * (v + H[off]);    // v + (x - v)/tau, tau = 2
        const float s = (h >= 1.0f) ? 1.0f : 0.0f;
        H[off] = s;
        v = h * (1.0f - s);
    }
}

extern "C" void kernel_launch(void* const* d_in, const int* in_sizes, int n_in,
                              void* d_out, int out_size, void* d_ws, size_t ws_size,
                              hipStream_t stream) {
    const float* x  = (const float*)d_in[0];   // [T, B, N]
    const float* W1 = (const float*)d_in[1];   // [N, N]
    const float* W2 = (const float*)d_in[2];   // [N, N]
    float* out = (float*)d_out;                // [T, B, N]
    float* H   = (float*)d_ws;                 // workspace: T*B*N floats (32 MB)

    dim3 grid(N_DIM / BN, M_ROWS / BM);        // (8, 64)
    dim3 block(256);

    gemm_f32_wmma<<<grid, block, 0, stream>>>(x, W1, H);     // H = X @ W1^T
    lif_scan<<<(B_DIM * N_DIM) / 256, block, 0, stream>>>(H); // spikes in place
    gemm_f32_wmma<<<grid, block, 0, stream>>>(H, W2, out);   // out = spikes @ W2^T
}